// GATLayer_37366215475783
// MI455X (gfx1250) — compile-verified
//
#include <hip/hip_runtime.h>
#include <cstdint>
#include <cstddef>

typedef __attribute__((ext_vector_type(2))) float v2f;
typedef __attribute__((ext_vector_type(8))) float v8f;

#define IN_DIM   128
#define OUT_DIM  128
#define HEADS    4
#define HEAD_DIM 32

// ---- order-preserving float <-> signed-int key (for atomicMax on int) ----
__device__ __forceinline__ int f2key(float f) {
  int i = __float_as_int(f);
  return i >= 0 ? i : (i ^ 0x7fffffff);
}
__device__ __forceinline__ float key2f(int k) {
  int i = k >= 0 ? k : (k ^ 0x7fffffff);
  return __int_as_float(i);
}

// ---------------------------------------------------------------------------
// Kernel 1: Wx[N,128] = x[N,128] @ W[128,128]^T  via V_WMMA_F32_16X16X4_F32
// Block = 256 threads (8 waves). Block handles one 16-node M-tile;
// wave w handles the 16-wide output tile [16w, 16w+16).
// A fragment (16x4 f32): lane<16 -> {x[m,k], x[m,k+1]}, lane>=16 -> K+2,K+3.
// B fragment (4x16 f32): symmetric, from W rows (W^T columns).
// C/D: 8 VGPRs, M = r + 8*(lane>=16), N = lane&15.
// ---------------------------------------------------------------------------
__global__ void gat_gemm_wmma(const float* __restrict__ x,
                              const float* __restrict__ W,
                              float* __restrict__ Wx, int N) {
  const int tile  = blockIdx.x;
  const int wave  = threadIdx.x >> 5;
  const int lane  = threadIdx.x & 31;
  const int mbase = tile * 16;
  const int obase = wave * 16;
  const int half  = lane >> 4;     // 0 -> K pair {0,1}, 1 -> K pair {2,3}
  const int l15   = lane & 15;

  int arow = mbase + l15;
  if (arow >= N) arow = N - 1;     // clamp (dup row; extra rows never stored)

  const float* aptr = x + (size_t)arow * IN_DIM + 2 * half;
  const float* bptr = W + (size_t)(obase + l15) * IN_DIM + 2 * half;

  v8f c = {};
  #pragma unroll
  for (int k = 0; k < IN_DIM; k += 4) {
    v2f af = *(const v2f*)(aptr + k);
    v2f bf = *(const v2f*)(bptr + k);
    c = __builtin_amdgcn_wmma_f32_16x16x4_f32(false, af, false, bf,
                                              (short)0, c, false, false);
  }

  #pragma unroll
  for (int r = 0; r < 8; ++r) {
    int row = mbase + r + 8 * half;
    if (row < N) Wx[(size_t)row * OUT_DIM + obase + l15] = c[r];
  }
}

// ---------------------------------------------------------------------------
// Kernel 2: per-node attention scores.
// s1[n,h] = dot(Wx[n, h*32 : h*32+32], a[0:32])
// s2[n,h] = dot(Wx[n, h*32 : h*32+32], a[32:64])
// ---------------------------------------------------------------------------
__global__ void gat_scores(const float* __restrict__ Wx,
                           const float* __restrict__ a,
                           float* __restrict__ s1, float* __restrict__ s2,
                           int N) {
  int t = blockIdx.x * blockDim.x + threadIdx.x;  // t = n*HEADS + h
  if (t >= N * HEADS) return;
  int n = t >> 2, h = t & 3;
  const float4* p  = (const float4*)(Wx + (size_t)n * OUT_DIM + h * HEAD_DIM);
  const float4* a1 = (const float4*)(a);
  const float4* a2 = (const float4*)(a + HEAD_DIM);
  float acc1 = 0.f, acc2 = 0.f;
  #pragma unroll
  for (int d = 0; d < HEAD_DIM / 4; ++d) {
    float4 v  = p[d];
    float4 c1 = a1[d];
    float4 c2 = a2[d];
    acc1 += v.x * c1.x + v.y * c1.y + v.z * c1.z + v.w * c1.w;
    acc2 += v.x * c2.x + v.y * c2.y + v.z * c2.z + v.w * c2.w;
  }
  s1[t] = acc1;
  s2[t] = acc2;
}

// ---------------------------------------------------------------------------
// Kernel 3: init accumulators (acc=0, denom=0, mkey=INT_MIN). Runs every call
// (harness does not re-poison between replays; this keeps us deterministic).
// ---------------------------------------------------------------------------
__global__ void gat_init(float* __restrict__ acc, float* __restrict__ denom,
                         int* __restrict__ mkey, int N) {
  int i = blockIdx.x * blockDim.x + threadIdx.x;
  if (i < N * OUT_DIM) acc[i] = 0.f;
  if (i < N * HEADS) { denom[i] = 0.f; mkey[i] = (int)0x80000000; }
}

// ---------------------------------------------------------------------------
// Kernel 4: segment max of leaky_relu(s1[src]+s2[dst]) by dst (per head),
// using i32 atomicMax on order-preserving keys.
// ---------------------------------------------------------------------------
__global__ void gat_edge_max(const int* __restrict__ ei,
                             const float* __restrict__ s1,
                             const float* __restrict__ s2,
                             int* __restrict__ mkey, int E) {
  int e = blockIdx.x * blockDim.x + threadIdx.x;
  if (e >= E) return;
  int src = ei[e], dst = ei[E + e];
  float4 v1 = *(const float4*)(s1 + (size_t)src * HEADS);
  float4 v2 = *(const float4*)(s2 + (size_t)dst * HEADS);
  float el[HEADS] = {v1.x + v2.x, v1.y + v2.y, v1.z + v2.z, v1.w + v2.w};
  #pragma unroll
  for (int h = 0; h < HEADS; ++h) {
    float v = el[h] > 0.f ? el[h] : 0.2f * el[h];
    atomicMax(&mkey[(size_t)dst * HEADS + h], f2key(v));
  }
}

// ---------------------------------------------------------------------------
// Kernel 5: ex[e,h] = exp(el - m[dst,h]); denom[dst,h] += ex (f32 atomic add).
// ---------------------------------------------------------------------------
__global__ void gat_edge_exp(const int* __restrict__ ei,
                             const float* __restrict__ s1,
                             const float* __restrict__ s2,
                             const int* __restrict__ mkey,
                             float* __restrict__ ex,
                             float* __restrict__ denom, int E) {
  int e = blockIdx.x * blockDim.x + threadIdx.x;
  if (e >= E) return;
  int src = ei[e], dst = ei[E + e];
  float4 v1 = *(const float4*)(s1 + (size_t)src * HEADS);
  float4 v2 = *(const float4*)(s2 + (size_t)dst * HEADS);
  float el[HEADS] = {v1.x + v2.x, v1.y + v2.y, v1.z + v2.z, v1.w + v2.w};
  float4 exv;
  #pragma unroll
  for (int h = 0; h < HEADS; ++h) {
    float v = el[h] > 0.f ? el[h] : 0.2f * el[h];
    float m = key2f(mkey[(size_t)dst * HEADS + h]);
    float xv = __expf(v - m);
    ((float*)&exv)[h] = xv;
    unsafeAtomicAdd(&denom[(size_t)dst * HEADS + h], xv);
  }
  *(float4*)(ex + (size_t)e * HEADS) = exv;
}

// ---------------------------------------------------------------------------
// Kernel 5b: denom -> 1/denom  (one rcp per node*head instead of one divide
// per edge*dim in the aggregation pass: ~200K rcps vs ~102M divides).
// Nodes with no in-edges produce inf but are never referenced by any edge.
// ---------------------------------------------------------------------------
__global__ void gat_recip(float* __restrict__ denom, int total) {
  int i = blockIdx.x * blockDim.x + threadIdx.x;
  if (i >= total) return;
  denom[i] = __frcp_rn(denom[i]);
}

// ---------------------------------------------------------------------------
// Kernel 6: acc[dst, j] += (ex[e,h] * rden[dst,h]) * Wx[src, j].
// One lane per (edge, 4 consecutive dims): float4 gather of Wx[src]
// (global_load_b128), 4 f32 global atomic adds into L2-resident acc.
// E*32 threads total.
// ---------------------------------------------------------------------------
__global__ void gat_edge_aggr(const int* __restrict__ ei,
                              const float* __restrict__ Wx,
                              const float* __restrict__ ex,
                              const float* __restrict__ rden,
                              float* __restrict__ acc, int E) {
  long long gid = (long long)blockIdx.x * blockDim.x + threadIdx.x;
  int e  = (int)(gid >> 5);
  int q  = (int)(gid & 31);        // quad index: dims [4q, 4q+4)
  if (e >= E) return;
  int src = ei[e], dst = ei[E + e];
  int j = q << 2;
  int h = j >> 5;                  // = q >> 3
  float alpha = ex[(size_t)e * HEADS + h] * rden[(size_t)dst * HEADS + h];
  float4 wv = *(const float4*)(Wx + (size_t)src * OUT_DIM + j);
  float* ap = acc + (size_t)dst * OUT_DIM + j;
  unsafeAtomicAdd(ap + 0, alpha * wv.x);
  unsafeAtomicAdd(ap + 1, alpha * wv.y);
  unsafeAtomicAdd(ap + 2, alpha * wv.z);
  unsafeAtomicAdd(ap + 3, alpha * wv.w);
}

// ---------------------------------------------------------------------------
// Kernel 7: out = elu(acc)
// ---------------------------------------------------------------------------
__global__ void gat_elu(const float* __restrict__ acc, float* __restrict__ out,
                        int total) {
  int i = blockIdx.x * blockDim.x + threadIdx.x;
  if (i >= total) return;
  float v = acc[i];
  out[i] = v > 0.f ? v : (__expf(v) - 1.0f);
}

// ---------------------------------------------------------------------------
extern "C" void kernel_launch(void* const* d_in, const int* in_sizes, int n_in,
                              void* d_out, int out_size, void* d_ws, size_t ws_size,
                              hipStream_t stream) {
  const float* x  = (const float*)d_in[0];
  const float* W  = (const float*)d_in[1];
  const float* a  = (const float*)d_in[2];
  const int*   ei = (const int*)d_in[3];
  const int N = in_sizes[0] / IN_DIM;
  const int E = in_sizes[3] / 2;
  float* out = (float*)d_out;

  // workspace carve-out (256B aligned slabs)
  char* ws = (char*)d_ws;
  size_t off = 0;
  auto carve = [&](size_t bytes) -> void* {
    void* p = ws + off;
    off += (bytes + 255) & ~(size_t)255;
    return p;
  };
  float* Wx    = (float*)carve((size_t)N * OUT_DIM * sizeof(float)); // 25.6 MB
  float* s1    = (float*)carve((size_t)N * HEADS * sizeof(float));
  float* s2    = (float*)carve((size_t)N * HEADS * sizeof(float));
  int*   mkey  = (int*)  carve((size_t)N * HEADS * sizeof(int));
  float* denom = (float*)carve((size_t)N * HEADS * sizeof(float));
  float* ex    = (float*)carve((size_t)E * HEADS * sizeof(float));   // 12.8 MB
  float* acc   = (float*)carve((size_t)N * OUT_DIM * sizeof(float)); // 25.6 MB
  (void)ws_size;

  // 1) GEMM (WMMA f32): one 16-node tile per block, 8 waves cover 128 outputs
  gat_gemm_wmma<<<(N + 15) / 16, 256, 0, stream>>>(x, W, Wx, N);

  // 2) per-node scores
  gat_scores<<<(N * HEADS + 255) / 256, 256, 0, stream>>>(Wx, a, s1, s2, N);

  // 3) init accumulators
  gat_init<<<(N * OUT_DIM + 255) / 256, 256, 0, stream>>>(acc, denom, mkey, N);

  // 4) segment max
  gat_edge_max<<<(E + 255) / 256, 256, 0, stream>>>(ei, s1, s2, mkey, E);

  // 5) exp + denominator
  gat_edge_exp<<<(E + 255) / 256, 256, 0, stream>>>(ei, s1, s2, mkey, ex, denom, E);

  // 5b) denom -> 1/denom
  gat_recip<<<(N * HEADS + 255) / 256, 256, 0, stream>>>(denom, N * HEADS);

  // 6) weighted scatter-aggregate (one lane per edge-quad, float4 gathers)
  {
    long long threads = (long long)E * 32;
    int blocks = (int)((threads + 255) / 256);
    gat_edge_aggr<<<blocks, 256, 0, stream>>>(ei, Wx, ex, denom, acc, E);
  }

  // 7) ELU -> output
  gat_elu<<<(N * OUT_DIM + 255) / 256, 256, 0, stream>>>(acc, out, N * OUT_DIM);
}